// GCNModel_59227599011868
// MI455X (gfx1250) — compile-verified
//
#include <hip/hip_runtime.h>
#include <math.h>

typedef __attribute__((ext_vector_type(2))) float v2f;
typedef __attribute__((ext_vector_type(8))) float v8f;

#define IN_DIM  128
#define HID_DIM 128
#define OUT_DIM 64

// ----------------- degree / norm -----------------
__global__ void k_deg_init(float* deg, int n) {
    int i = blockIdx.x * blockDim.x + threadIdx.x;
    if (i < n) deg[i] = 1.0f;  // self-loop contributes 1
}

__global__ void k_deg_edges(const int* __restrict__ dst, float* deg, int e) {
    int i = blockIdx.x * blockDim.x + threadIdx.x;
    if (i < e) atomicAdd(&deg[dst[i]], 1.0f);
}

__global__ void k_deg_to_dinv(float* deg, int n) {
    int i = blockIdx.x * blockDim.x + threadIdx.x;
    if (i < n) {
        float d = deg[i];
        deg[i] = (d > 0.0f) ? rsqrtf(d) : 0.0f;
    }
}

// ----------------- fp32 WMMA GEMM: Out[nRows, NT*16] = A[nRows,128] @ W[128, NT*16] -----
// One wave computes a 16-row tile across all NT column tiles, K=128 in steps of 4
// via V_WMMA_F32_16X16X4_F32 (full fp32 precision, matches fp32 reference).
template<int NT>
__global__ void __launch_bounds__(256)
k_gemm_wmma(const float* __restrict__ A, const float* __restrict__ W,
            float* __restrict__ Out, int nRows)
{
    constexpr int OUTD = NT * 16;
    const int lane = threadIdx.x & 31;
    const int wave = threadIdx.x >> 5;
    const int tile = blockIdx.x * (blockDim.x >> 5) + wave;
    const int row0 = tile * 16;
    if (row0 >= nRows) return;          // wave-uniform: EXEC stays all-1s for WMMA

    const int half = lane >> 4;         // 0: lanes 0-15 (K=0,1), 1: lanes 16-31 (K=2,3)
    const int l16  = lane & 15;

    int arow = row0 + l16;
    if (arow >= nRows) arow = nRows - 1;  // clamp (stores are guarded)
    const float* ap = A + (size_t)arow * IN_DIM;

    v8f zero = {};
    v8f acc[NT];
#pragma unroll
    for (int t = 0; t < NT; ++t) acc[t] = zero;

    for (int k = 0; k < IN_DIM; k += 4) {
        const int kk = k + (half << 1);
        v2f a;
        a.x = ap[kk];
        a.y = ap[kk + 1];
#pragma unroll
        for (int t = 0; t < NT; ++t) {
            const int col = t * 16 + l16;
            v2f b;
            b.x = W[(size_t)kk * OUTD + col];
            b.y = W[(size_t)(kk + 1) * OUTD + col];
            // 8 args: (neg_a, A, neg_b, B, c_mod, C, reuse_a, reuse_b)
            acc[t] = __builtin_amdgcn_wmma_f32_16x16x4_f32(
                false, a, false, b, (short)0, acc[t], false, false);
        }
    }

#pragma unroll
    for (int t = 0; t < NT; ++t) {
#pragma unroll
        for (int v = 0; v < 8; ++v) {
            const int m = v + (half << 3);   // D row = v + 8*(lane>=16)
            const int r = row0 + m;
            if (r < nRows)
                Out[(size_t)r * OUTD + t * 16 + l16] = acc[t][v];
        }
    }
}

// ----------------- edge scatter: agg[dst] += h[src] * dinv[src]*dinv[dst] -----------------
template<int D>
__global__ void k_scatter(const int* __restrict__ src, const int* __restrict__ dst,
                          const float* __restrict__ h, const float* __restrict__ dinv,
                          float* __restrict__ agg, int nEdges)
{
    constexpr int CH = D / 4;           // float4 chunks per row
    long long idx = (long long)blockIdx.x * blockDim.x + threadIdx.x;
    if (idx >= (long long)nEdges * CH) return;
    const int e = (int)(idx / CH);
    const int c = (int)(idx % CH);
    const int s = src[e];
    const int d = dst[e];
    const float norm = dinv[s] * dinv[d];
    const float4 m = *((const float4*)(h + (size_t)s * D) + c);
    float* op = agg + (size_t)d * D + c * 4;
    atomicAdd(op + 0, m.x * norm);
    atomicAdd(op + 1, m.y * norm);
    atomicAdd(op + 2, m.z * norm);
    atomicAdd(op + 3, m.w * norm);
}

// ----------------- finalize: agg = ELU(agg + h*dinv^2 (self-loop) + bias) -----------------
template<int D>
__global__ void k_finalize(float* __restrict__ agg, const float* __restrict__ h,
                           const float* __restrict__ dinv, const float* __restrict__ bias,
                           int nNodes)
{
    long long idx = (long long)blockIdx.x * blockDim.x + threadIdx.x;
    if (idx >= (long long)nNodes * D) return;
    const int i = (int)(idx / D);
    const int j = (int)(idx % D);
    const float di = dinv[i];
    const float v = agg[idx] + h[idx] * (di * di) + bias[j];
    agg[idx] = (v > 0.0f) ? v : expm1f(v);   // ELU, alpha = 1
}

extern "C" void kernel_launch(void* const* d_in, const int* in_sizes, int n_in,
                              void* d_out, int out_size, void* d_ws, size_t ws_size,
                              hipStream_t stream)
{
    (void)n_in; (void)out_size; (void)ws_size;
    const float* x  = (const float*)d_in[0];
    const int*   ei = (const int*)d_in[1];
    const float* W1 = (const float*)d_in[2];
    const float* b1 = (const float*)d_in[3];
    const float* W2 = (const float*)d_in[4];
    const float* b2 = (const float*)d_in[5];

    const int N = in_sizes[0] / IN_DIM;   // 100000
    const int E = in_sizes[1] / 2;        // 1600000
    const int* src = ei;                  // edge_index[0]
    const int* dst = ei + E;              // edge_index[1]

    // workspace carve-out (256B aligned): dinv[N], h1[N*128] (reused as h2), agg1[N*128]
    char* ws = (char*)d_ws;
    size_t off = 0;
    auto carve = [&](size_t bytes) {
        char* p = ws + off;
        off = (off + bytes + 255) & ~(size_t)255;
        return p;
    };
    float* dinv = (float*)carve((size_t)N * sizeof(float));
    float* h1   = (float*)carve((size_t)N * HID_DIM * sizeof(float));
    float* agg1 = (float*)carve((size_t)N * HID_DIM * sizeof(float));
    float* h2   = h1;                     // h1 dead after finalize1 -> reuse for GEMM2 output
    float* agg2 = (float*)d_out;          // aggregate layer 2 directly into d_out

    const int TB = 256;

    // normalization coefficients
    k_deg_init   <<<(N + TB - 1) / TB, TB, 0, stream>>>(dinv, N);
    k_deg_edges  <<<(E + TB - 1) / TB, TB, 0, stream>>>(dst, dinv, E);
    k_deg_to_dinv<<<(N + TB - 1) / TB, TB, 0, stream>>>(dinv, N);

    const int tiles   = (N + 15) / 16;
    const int gblocks = (tiles + 7) / 8;   // 8 waves (256 threads) per block

    // ---- layer 1: h1 = x @ W1; agg1 = Ahat h1; elu(+b1) ----
    k_gemm_wmma<HID_DIM / 16><<<gblocks, TB, 0, stream>>>(x, W1, h1, N);
    hipMemsetAsync(agg1, 0, (size_t)N * HID_DIM * sizeof(float), stream);
    {
        long long tot = (long long)E * (HID_DIM / 4);
        k_scatter<HID_DIM><<<(unsigned)((tot + TB - 1) / TB), TB, 0, stream>>>(
            src, dst, h1, dinv, agg1, E);
        long long fn = (long long)N * HID_DIM;
        k_finalize<HID_DIM><<<(unsigned)((fn + TB - 1) / TB), TB, 0, stream>>>(
            agg1, h1, dinv, b1, N);
    }

    // ---- layer 2: h2 = agg1 @ W2; d_out = Ahat h2; elu(+b2) ----
    k_gemm_wmma<OUT_DIM / 16><<<gblocks, TB, 0, stream>>>(agg1, W2, h2, N);
    hipMemsetAsync(agg2, 0, (size_t)N * OUT_DIM * sizeof(float), stream);
    {
        long long tot = (long long)E * (OUT_DIM / 4);
        k_scatter<OUT_DIM><<<(unsigned)((tot + TB - 1) / TB), TB, 0, stream>>>(
            src, dst, h2, dinv, agg2, E);
        long long fn = (long long)N * OUT_DIM;
        k_finalize<OUT_DIM><<<(unsigned)((fn + TB - 1) / TB), TB, 0, stream>>>(
            agg2, h2, dinv, b2, N);
    }
}